// _PatchCausalEMAEncoder_6167573037289
// MI455X (gfx1250) — compile-verified
//
#include <hip/hip_runtime.h>
#include <cstdint>
#include <cstddef>

// ---------------------------------------------------------------------------
// PatchCausalEMAEncoder final-state kernel for MI455X (gfx1250).
//
//   out[b,h] = h_P  where  h_t = a_h * h_{t-1} + (1-a_h) * u[b,t,h],
//   a = sigmoid(a_logit),  u: (2048, 128, 512) f32,  out: (2048, 512) f32.
//
// Memory-bound (0.52 FLOP/byte): stream 512 MB of u once at HBM rate.
// CDNA5 path: double-buffered GLOBAL_LOAD_ASYNC_TO_LDS_B128 + s_wait_asynccnt.
// ---------------------------------------------------------------------------

namespace {
constexpr int kH       = 512;            // HIDDEN_DIM
constexpr int kP       = 128;            // scan length
constexpr int kBC      = 2048;           // batch rows
constexpr int kTileT   = 8;              // time rows per LDS tile (16 KB)
constexpr int kNumTiles = kP / kTileT;   // 16
constexpr int kThreads = kH / 4;         // 128 threads: one float4 per 4 channels
}

#if defined(__AMDGCN__) &&                                              \
    __has_builtin(__builtin_amdgcn_global_load_async_to_lds_b128) &&    \
    __has_builtin(__builtin_amdgcn_s_wait_asynccnt)
#define EMA_USE_ASYNC 1
#else
#define EMA_USE_ASYNC 0
#endif

// Builtin prototype (from clang diagnostic): param 0 is
//   'int __vector(4) __device__ *'  i.e. v4i32 in address_space(1).
typedef int ema_v4i __attribute__((__vector_size__(16)));
typedef __attribute__((address_space(1))) ema_v4i ema_gvec;  // global v4i32
typedef __attribute__((address_space(3))) ema_v4i ema_lvec;  // LDS    v4i32

__global__ __launch_bounds__(kThreads) void ema_final_state_kernel(
    const float* __restrict__ u,        // (BC, P, H)
    const float* __restrict__ a_logit,  // (H,)
    float* __restrict__ out)            // (BC, H)
{
  const int tid = (int)threadIdx.x;
  const int b   = (int)blockIdx.x;
  const int c   = tid << 2;             // first of this thread's 4 channels

  // Per-channel smoothing factors (computed once; tiny vs. 256 KB stream/block)
  const float4 al = *reinterpret_cast<const float4*>(a_logit + c);
  float4 a, oma;
  a.x = 1.0f / (1.0f + __expf(-al.x));
  a.y = 1.0f / (1.0f + __expf(-al.y));
  a.z = 1.0f / (1.0f + __expf(-al.z));
  a.w = 1.0f / (1.0f + __expf(-al.w));
  oma.x = 1.0f - a.x;
  oma.y = 1.0f - a.y;
  oma.z = 1.0f - a.z;
  oma.w = 1.0f - a.w;

  const float* ub = u + (size_t)b * kP * kH;
  float4 acc = make_float4(0.0f, 0.0f, 0.0f, 0.0f);

#if EMA_USE_ASYNC
  // Double-buffered staging: 2 x (8 rows x 2 KB) = 32 KB LDS.
  // Each lane async-copies the exact 16 B/row it will later consume, so
  // per-wave ASYNCcnt ordering is sufficient -- no workgroup barrier needed.
  __shared__ float lds[2][kTileT][kH];

  // Prologue: tile 0 -> buffer 0 (8 async ops outstanding per wave).
  {
    const float* g0 = ub + c;
    float*       l0 = &lds[0][0][c];
#pragma unroll
    for (int r = 0; r < kTileT; ++r) {
      __builtin_amdgcn_global_load_async_to_lds_b128(
          (ema_gvec*)(g0 + (size_t)r * kH),
          (ema_lvec*)(l0 + (size_t)r * kH),
          /*offset=*/0, /*cpol=*/0);
    }
  }

  for (int k = 0; k < kNumTiles; ++k) {
    if (k + 1 < kNumTiles) {
      // Kick off tile k+1 into the other buffer, then wait for tile k.
      const float* gn = ub + (size_t)(k + 1) * kTileT * kH + c;
      float*       ln = &lds[(k + 1) & 1][0][c];
#pragma unroll
      for (int r = 0; r < kTileT; ++r) {
        __builtin_amdgcn_global_load_async_to_lds_b128(
            (ema_gvec*)(gn + (size_t)r * kH),
            (ema_lvec*)(ln + (size_t)r * kH),
            /*offset=*/0, /*cpol=*/0);
      }
      // Async loads complete in order: <= kTileT outstanding => tile k landed.
      __builtin_amdgcn_s_wait_asynccnt(kTileT);
    } else {
      __builtin_amdgcn_s_wait_asynccnt(0);  // drain last tile
    }

    const float* lb = &lds[k & 1][0][c];
#pragma unroll
    for (int r = 0; r < kTileT; ++r) {
      const float4 uv = *reinterpret_cast<const float4*>(lb + (size_t)r * kH);
      acc.x = a.x * acc.x + oma.x * uv.x;
      acc.y = a.y * acc.y + oma.y * uv.y;
      acc.z = a.z * acc.z + oma.z * uv.z;
      acc.w = a.w * acc.w + oma.w * uv.w;
    }
  }
#else
  // Fallback: direct coalesced streaming loads + software prefetch
  // (emits global_prefetch_b8 on gfx1250).
#pragma unroll 8
  for (int t = 0; t < kP; ++t) {
    __builtin_prefetch(ub + (size_t)(t + 8) * kH + c, 0, 1);
    const float4 uv =
        *reinterpret_cast<const float4*>(ub + (size_t)t * kH + c);
    acc.x = a.x * acc.x + oma.x * uv.x;
    acc.y = a.y * acc.y + oma.y * uv.y;
    acc.z = a.z * acc.z + oma.z * uv.z;
    acc.w = a.w * acc.w + oma.w * uv.w;
  }
#endif

  *reinterpret_cast<float4*>(out + (size_t)b * kH + c) = acc;
}

// ---------------------------------------------------------------------------
// Harness entry point.
//   d_in[0]: u (BC*P*H f32), d_in[1]: a_logit (H f32), d_out: (BC*H f32)
// ---------------------------------------------------------------------------
extern "C" void kernel_launch(void* const* d_in, const int* in_sizes, int n_in,
                              void* d_out, int out_size, void* d_ws,
                              size_t ws_size, hipStream_t stream) {
  (void)in_sizes; (void)n_in; (void)out_size; (void)d_ws; (void)ws_size;
  const float* u       = (const float*)d_in[0];
  const float* a_logit = (const float*)d_in[1];
  float*       out     = (float*)d_out;

  dim3 grid(kBC);
  dim3 block(kThreads);
  hipLaunchKernelGGL(ema_final_state_kernel, grid, block, 0, stream,
                     u, a_logit, out);
}